// LengthRegulator_2448131359234
// MI455X (gfx1250) — compile-verified
//
#include <hip/hip_runtime.h>

// LengthRegulator for MI455X (gfx1250, wave32).
// B=16, T=512, D=384, DUR_MAX=8, MAX_OUT = T*(DUR_MAX-1) = 3584.
// Pure bandwidth problem (~101 MB HBM traffic -> ~4.3 us floor @ 23.3 TB/s).

#define B_       16
#define T_       512
#define D_       384
#define DUR_MAX_ 8
#define MAX_OUT_ (T_ * (DUR_MAX_ - 1))   // 3584
#define NROWS_   (B_ * MAX_OUT_)         // 57344
#define F4_PER_ROW_ (D_ / 4)             // 96 float4 per row
#define ROWS_PER_BLOCK_ 4                // 384 threads = 12 waves = 4 rows

typedef float v4f __attribute__((ext_vector_type(4)));
typedef int   v4i __attribute__((ext_vector_type(4)));

// ---------------------------------------------------------------------------
// Kernel 1: per-batch inclusive cumsum of dur[T_] using one wave32.
// Each lane owns 16 consecutive durations (4x int4 loads), does an in-lane
// prefix sum, then a 5-step shfl_up wave scan of the lane totals.
// ---------------------------------------------------------------------------
__global__ __launch_bounds__(32) void lr_scan_kernel(const int* __restrict__ dur,
                                                     int* __restrict__ ends) {
    const int b    = blockIdx.x;      // one block (one wave) per batch
    const int lane = threadIdx.x;     // 0..31

    const v4i* dur4 = (const v4i*)(dur + b * T_);
    v4i q0 = dur4[lane * 4 + 0];
    v4i q1 = dur4[lane * 4 + 1];
    v4i q2 = dur4[lane * 4 + 2];
    v4i q3 = dur4[lane * 4 + 3];

    int vals[16] = { q0.x, q0.y, q0.z, q0.w,
                     q1.x, q1.y, q1.z, q1.w,
                     q2.x, q2.y, q2.z, q2.w,
                     q3.x, q3.y, q3.z, q3.w };

    int s[16];
    int acc = 0;
#pragma unroll
    for (int i = 0; i < 16; ++i) { acc += vals[i]; s[i] = acc; }

    // Inclusive wave32 scan of per-lane totals (Hillis-Steele via shfl_up).
    int incl = acc;
#pragma unroll
    for (int off = 1; off < 32; off <<= 1) {
        int n = __shfl_up(incl, off, 32);
        if (lane >= off) incl += n;
    }
    const int excl = incl - acc;      // exclusive prefix for this lane

    v4i* e4 = (v4i*)(ends + b * T_ + lane * 16);
    e4[0] = (v4i){ s[0]  + excl, s[1]  + excl, s[2]  + excl, s[3]  + excl };
    e4[1] = (v4i){ s[4]  + excl, s[5]  + excl, s[6]  + excl, s[7]  + excl };
    e4[2] = (v4i){ s[8]  + excl, s[9]  + excl, s[10] + excl, s[11] + excl };
    e4[3] = (v4i){ s[12] + excl, s[13] + excl, s[14] + excl, s[15] + excl };
}

// ---------------------------------------------------------------------------
// Kernel 2: expand/gather. One output row (384 floats = 96 float4) is exactly
// 3 wave32s, so every wave is wave-uniform in (batch, out-row). We exploit
// that with readfirstlane so the searchsorted binary search and the validity
// branch scalarize (SALU/SMEM), and x loads use a uniform base + lane offset.
// Output is write-once -> non-temporal b128 stores (don't evict x from L2).
// ---------------------------------------------------------------------------
__global__ __launch_bounds__(ROWS_PER_BLOCK_ * F4_PER_ROW_)
void lr_gather_kernel(const float* __restrict__ x,
                      const int* __restrict__ ends,
                      float* __restrict__ out) {
    const int row = blockIdx.x * ROWS_PER_BLOCK_ + threadIdx.x / F4_PER_ROW_;
    const int col = threadIdx.x % F4_PER_ROW_;      // 0..95, varies per lane
    if (row >= NROWS_) return;                       // whole wave exits together

    int b = row / MAX_OUT_;
    int j = row - b * MAX_OUT_;
    // Wave-uniform by construction; make it explicit for the scalarizer.
    b = __builtin_amdgcn_readfirstlane(b);
    j = __builtin_amdgcn_readfirstlane(j);

    const int* eb    = ends + b * T_;
    const int  total = eb[T_ - 1];

    v4f v = (v4f){ 0.f, 0.f, 0.f, 0.f };
    if (j < total) {                                 // uniform branch
        // searchsorted(ends, j, side='right'): first idx with ends[idx] > j.
        int lo = 0, hi = T_;
#pragma unroll 1
        while (lo < hi) {
            const int mid = (lo + hi) >> 1;
            if (eb[mid] <= j) lo = mid + 1; else hi = mid;
        }
        const int src = __builtin_amdgcn_readfirstlane(lo);  // < T_ since j < total
        const v4f* xrow = (const v4f*)(x + (size_t)(b * T_ + src) * D_);
        v = xrow[col];                               // global_load_b128, coalesced
    }

    v4f* orow = (v4f*)(out + (size_t)row * D_);
    __builtin_nontemporal_store(v, orow + col);      // b128 store, TH=NT
}

// ---------------------------------------------------------------------------
extern "C" void kernel_launch(void* const* d_in, const int* in_sizes, int n_in,
                              void* d_out, int out_size, void* d_ws, size_t ws_size,
                              hipStream_t stream) {
    const float* x   = (const float*)d_in[0];   // (B, T, D) float32
    const int*   dur = (const int*)d_in[1];     // (B, T)    int32
    float*       out = (float*)d_out;           // (B, MAX_OUT, D) float32
    int*         ends = (int*)d_ws;             // B*T ints = 32 KB scratch

    lr_scan_kernel<<<B_, 32, 0, stream>>>(dur, ends);

    const int grid = NROWS_ / ROWS_PER_BLOCK_;  // 57344 / 4 = 14336
    lr_gather_kernel<<<grid, ROWS_PER_BLOCK_ * F4_PER_ROW_, 0, stream>>>(x, ends, out);
}